// CustomLoss_31714038514005
// MI455X (gfx1250) — compile-verified
//
#include <hip/hip_runtime.h>

typedef __attribute__((ext_vector_type(2))) float v2f;
typedef __attribute__((ext_vector_type(4))) float f4;
typedef __attribute__((ext_vector_type(8))) float v8f;

#define N_ELEM   33554432u            // 2^25
#define N_VEC4   (N_ELEM / 4u)        // 2^23 float4s
#define BLOCKS   2048u
#define THREADS  256u                 // 8 wave32 waves per block
#define STRIDE   (BLOCKS * THREADS)   // 524288
#define ITERS    (N_VEC4 / STRIDE)    // exactly 16 iterations per thread

// Wave-level sum of 32 per-lane partials using V_WMMA_F32_16X16X4_F32.
// A (16x4 f32, 2 VGPRs/lane) carries each lane's partial in .x and 0 in .y;
// B (4x16 f32) is all-ones. D[i][j] = sum_k A[i][k]; summing the 8 D VGPRs
// per lane leaves lanes 0-15 with sum(rows 0..7) and lanes 16-31 with
// sum(rows 8..15), replicated across columns. One shfl_xor(16) add -> total.
// EXEC must be all 1s at the call site (ISA 7.12 restriction).
__device__ __forceinline__ float wave_sum_wmma(float s) {
    v2f a; a.x = s;    a.y = 0.0f;
    v2f b; b.x = 1.0f; b.y = 1.0f;
    v8f c = {};
    c = __builtin_amdgcn_wmma_f32_16x16x4_f32(
            /*neg_a=*/false, a, /*neg_b=*/false, b,
            /*c_mod=*/(short)0, c, /*reuse_a=*/false, /*reuse_b=*/false);
    float t = c[0] + c[1] + c[2] + c[3] + c[4] + c[5] + c[6] + c[7];
    t += __shfl_xor(t, 16, 32);
    return t;   // every lane holds the wave total
}

__device__ __forceinline__ float sq_acc(f4 a, f4 b, float s) {
    float dx = __builtin_fabsf(a.x - b.x);
    float dy = __builtin_fabsf(a.y - b.y);
    float dz = __builtin_fabsf(a.z - b.z);
    float dw = __builtin_fabsf(a.w - b.w);
    s = __builtin_fmaf(dx, dx, s);
    s = __builtin_fmaf(dy, dy, s);
    s = __builtin_fmaf(dz, dz, s);
    s = __builtin_fmaf(dw, dw, s);
    return s;
}

__global__ void __launch_bounds__(THREADS)
loss_partial_kernel(const f4* __restrict__ in, const f4* __restrict__ tg,
                    float* __restrict__ block_partials) {
    __shared__ float wave_sums[THREADS / 32];

    const unsigned gid = blockIdx.x * THREADS + threadIdx.x;
    float s = 0.0f;

    // ---- peeled iteration 0: only place the element-0 quirk can occur ----
    {
        f4 a = __builtin_nontemporal_load(&in[gid]);
        f4 b = __builtin_nontemporal_load(&tg[gid]);
        float dx = __builtin_fabsf(a.x - b.x);
        float dy = __builtin_fabsf(a.y - b.y);
        float dz = __builtin_fabsf(a.z - b.z);
        float dw = __builtin_fabsf(a.w - b.w);
        if (gid == 0u) {
            // reference quirk: conditionally scale element 0 before squaring
            bool hit = (dx == 3.0f) | (dx == 4.0f) | (dx == 5.0f) | (dx == 6.0f);
            dx = hit ? dx * 0.8f : dx;
        }
        s = __builtin_fmaf(dx, dx, s);
        s = __builtin_fmaf(dy, dy, s);
        s = __builtin_fmaf(dz, dz, s);
        s = __builtin_fmaf(dw, dw, s);
    }

    // ---- branch-free streaming loop: b128 non-temporal loads ----
    // (working set 256 MB > 192 MB L2, no reuse -> NT temporal hint)
    #pragma unroll 5
    for (unsigned k = 1; k < ITERS; ++k) {
        const unsigned i = gid + k * STRIDE;
        f4 a = __builtin_nontemporal_load(&in[i]);
        f4 b = __builtin_nontemporal_load(&tg[i]);
        s = sq_acc(a, b, s);
    }

    // Uniform trip counts -> fully converged (EXEC all ones) here.
    float wtot = wave_sum_wmma(s);

    const unsigned lane = threadIdx.x & 31u;
    const unsigned wave = threadIdx.x >> 5u;
    if (lane == 0u) wave_sums[wave] = wtot;
    __syncthreads();

    if (threadIdx.x == 0u) {
        float bsum = 0.0f;
        #pragma unroll
        for (int w = 0; w < (int)(THREADS / 32); ++w) bsum += wave_sums[w];
        block_partials[blockIdx.x] = bsum;
    }
}

__global__ void __launch_bounds__(THREADS)
loss_final_kernel(const float* __restrict__ block_partials, float* __restrict__ out) {
    __shared__ float wave_sums[THREADS / 32];

    float s = 0.0f;
    // 2048 partials / 256 threads = 8 each; identical trip counts.
    #pragma unroll
    for (unsigned k = 0; k < BLOCKS / THREADS; ++k) {
        s += block_partials[threadIdx.x + k * THREADS];
    }

    float wtot = wave_sum_wmma(s);

    const unsigned lane = threadIdx.x & 31u;
    const unsigned wave = threadIdx.x >> 5u;
    if (lane == 0u) wave_sums[wave] = wtot;
    __syncthreads();

    if (threadIdx.x == 0u) {
        float total = 0.0f;
        #pragma unroll
        for (int w = 0; w < (int)(THREADS / 32); ++w) total += wave_sums[w];
        // mean: divide by 2^25 (exact in fp32)
        out[0] = total * (1.0f / 33554432.0f);
    }
}

extern "C" void kernel_launch(void* const* d_in, const int* in_sizes, int n_in,
                              void* d_out, int out_size, void* d_ws, size_t ws_size,
                              hipStream_t stream) {
    const f4* in = (const f4*)d_in[0];
    const f4* tg = (const f4*)d_in[1];
    float* partials = (float*)d_ws;          // 2048 floats = 8 KB scratch
    float* out = (float*)d_out;

    loss_partial_kernel<<<BLOCKS, THREADS, 0, stream>>>(in, tg, partials);
    loss_final_kernel<<<1, THREADS, 0, stream>>>(partials, out);
}